// ClusterDiceLoss_2800318677064
// MI455X (gfx1250) — compile-verified
//
#include <hip/hip_runtime.h>
#include <stdint.h>

#ifndef __has_builtin
#define __has_builtin(x) 0
#endif

#if __has_builtin(__builtin_amdgcn_global_load_async_to_lds_b32) && \
    __has_builtin(__builtin_amdgcn_s_wait_asynccnt)
#define USE_ASYNC_LDS 1
#else
#define USE_ASYNC_LDS 0
#endif

#define AS1 __attribute__((address_space(1)))
#define AS3 __attribute__((address_space(3)))

static constexpr int Hc = 1024, Wc = 1024, Nc = Hc * Wc, Bc = 8;
static constexpr int TILE = 64, TPB = 256, PPT = (TILE * TILE) / TPB; // 16 px/thread
static constexpr int TILES_X = Wc / TILE, TILES_Y = Hc / TILE;        // 16 x 16
static constexpr float EPSF = 1e-6f;

// ---------- union-find on global labels (root = min flat index) ----------
__device__ __forceinline__ int findRoot(const int* __restrict__ L, int x) {
  int p = L[x];
  while (p != x) { x = p; p = L[x]; }
  return x;
}

__device__ __forceinline__ void unite(int* L, int a, int b) {
  while (true) {
    a = findRoot(L, a);
    b = findRoot(L, b);
    if (a == b) return;
    int mn = a < b ? a : b;
    int mx = a < b ? b : a;
    int old = atomicMin(&L[mx], mn);
    if (old == mx) return;
    a = old; b = mn;
  }
}

// ---------- kernel 1: per-tile CCL in LDS + init of accumulators ----------
__global__ __launch_bounds__(TPB)
void tile_ccl_kernel(const float* __restrict__ pred, const float* __restrict__ target,
                     int* __restrict__ L, unsigned char* __restrict__ Mk,
                     float* __restrict__ inter, float* __restrict__ uni,
                     float* __restrict__ cnt)
{
  __shared__ unsigned short lbl[TILE * TILE]; // local labels, 0xFFFF = unmasked
  __shared__ int s_changed;
#if USE_ASYNC_LDS
  __shared__ float s_p[TILE * TILE];
  __shared__ float s_t[TILE * TILE];
#endif
  const int tid = threadIdx.x;
  const int tileId = blockIdx.x;
  const int tx0 = (tileId % TILES_X) * TILE;
  const int ty0 = (tileId / TILES_X) * TILE;

  unsigned maskbits = 0;

#if USE_ASYNC_LDS
  // stage pred/target tile through LDS with gfx1250 async loads
  for (int k = 0; k < PPT; ++k) {
    int li = tid + k * TPB;
    int g = (ty0 + li / TILE) * Wc + tx0 + (li % TILE);
    __builtin_amdgcn_global_load_async_to_lds_b32(
        (AS1 int*)(pred + g), (AS3 int*)(&s_p[li]), 0, 0);
    __builtin_amdgcn_global_load_async_to_lds_b32(
        (AS1 int*)(target + g), (AS3 int*)(&s_t[li]), 0, 0);
  }
  __builtin_amdgcn_s_wait_asynccnt(0);
#endif

  for (int k = 0; k < PPT; ++k) {
    int li = tid + k * TPB;
    int g = (ty0 + li / TILE) * Wc + tx0 + (li % TILE);
#if USE_ASYNC_LDS
    float pv = s_p[li], tv = s_t[li];
#else
    float pv = pred[g], tv = target[g];
#endif
    bool m = (pv + tv) > 0.0f;
    lbl[li] = m ? (unsigned short)li : (unsigned short)0xFFFF;
    maskbits |= (m ? 1u : 0u) << k;
    // zero the per-root accumulators (tile covers a disjoint slice of [0,N))
    inter[g] = 0.0f; uni[g] = 0.0f; cnt[g] = 0.0f;
  }
  __syncthreads();

  // iterative neighbor-min + pointer doubling until fixed point
  for (int iter = 0; iter < 4096; ++iter) {
    if (tid == 0) s_changed = 0;
    __syncthreads();

    int nl[PPT];
    for (int k = 0; k < PPT; ++k) {
      int li = tid + k * TPB;
      int v = 0xFFFF;
      if ((maskbits >> k) & 1) {
        int lx = li % TILE, ly = li / TILE;
        v = lbl[li];
        if (lx > 0)        { int q = lbl[li - 1];    v = q < v ? q : v; }
        if (lx < TILE - 1) { int q = lbl[li + 1];    v = q < v ? q : v; }
        if (ly > 0)        { int q = lbl[li - TILE]; v = q < v ? q : v; }
        if (ly < TILE - 1) { int q = lbl[li + TILE]; v = q < v ? q : v; }
      }
      nl[k] = v;
    }
    __syncthreads();

    bool any = false;
    for (int k = 0; k < PPT; ++k) {
      int li = tid + k * TPB;
      if (((maskbits >> k) & 1) && nl[k] < (int)lbl[li]) {
        lbl[li] = (unsigned short)nl[k];
        any = true;
      }
    }
    if (any) s_changed = 1;
    __syncthreads();

    // two rounds of pointer doubling (path compression)
    for (int r = 0; r < 2; ++r) {
      for (int k = 0; k < PPT; ++k) {
        int li = tid + k * TPB;
        nl[k] = ((maskbits >> k) & 1) ? (int)lbl[lbl[li]] : 0xFFFF;
      }
      __syncthreads();
      for (int k = 0; k < PPT; ++k) {
        int li = tid + k * TPB;
        if ((maskbits >> k) & 1) lbl[li] = (unsigned short)nl[k];
      }
      __syncthreads();
    }

    if (!s_changed) break;
  }

  // write global labels (tile-local root -> global flat index) + mask
  for (int k = 0; k < PPT; ++k) {
    int li = tid + k * TPB;
    int g = (ty0 + li / TILE) * Wc + tx0 + (li % TILE);
    if ((maskbits >> k) & 1) {
      int r = lbl[li];
      int groot = (ty0 + r / TILE) * Wc + tx0 + (r % TILE);
      L[g] = groot;
      Mk[g] = 1;
    } else {
      L[g] = g;
      Mk[g] = 0;
    }
  }
}

// ---------- kernel 2: merge components across tile boundaries ----------
__global__ __launch_bounds__(256)
void merge_kernel(int* __restrict__ L, const unsigned char* __restrict__ Mk)
{
  const int PV = (TILES_X - 1) * Hc; // vertical tile boundaries: pairs (x, x+1)
  const int PH = (TILES_Y - 1) * Wc; // horizontal tile boundaries: pairs (y, y+1)
  int t = blockIdx.x * blockDim.x + threadIdx.x;
  if (t < PV) {
    int bidx = t / Hc;
    int row = t % Hc;
    int x = bidx * TILE + (TILE - 1);
    int g1 = row * Wc + x;
    int g2 = g1 + 1;
    if (Mk[g1] && Mk[g2]) unite(L, g1, g2);
  } else if (t < PV + PH) {
    int i = t - PV;
    int bidx = i / Wc;
    int col = i % Wc;
    int y = bidx * TILE + (TILE - 1);
    int g1 = y * Wc + col;
    int g2 = g1 + Wc;
    if (Mk[g1] && Mk[g2]) unite(L, g1, g2);
  }
}

// ---------- kernel 3: segmented accumulation with run aggregation ----------
__global__ __launch_bounds__(256)
void accum_kernel(const float* __restrict__ pred, const float* __restrict__ target,
                  int* __restrict__ L, const unsigned char* __restrict__ Mk,
                  float* __restrict__ inter, float* __restrict__ uni,
                  float* __restrict__ cnt)
{
  __shared__ int s_root[256];
  __shared__ float s_i[256], s_u[256];
  int tid = threadIdx.x;
  int i = blockIdx.x * 256 + tid;

  if (i + 65536 < Nc) {
    __builtin_prefetch(pred + i + 65536, 0, 0);   // global_prefetch_b8
    __builtin_prefetch(target + i + 65536, 0, 0);
  }

  int r = -1;
  float vi = 0.0f, vu = 0.0f;
  if (Mk[i]) {
    float p = pred[i], t = target[i];
    vi = p * t;
    vu = p + t;
    r = findRoot(L, i);
  }
  s_root[tid] = r; s_i[tid] = vi; s_u[tid] = vu;
  __syncthreads();

  // leader of each contiguous same-root run issues the atomics
  if (r >= 0 && (tid == 0 || s_root[tid - 1] != r)) {
    float A = 0.0f, Bv = 0.0f, C = 0.0f;
    int j = tid;
    while (j < 256 && s_root[j] == r) { A += s_i[j]; Bv += s_u[j]; C += 1.0f; ++j; }
    atomicAdd(&inter[r], A);
    atomicAdd(&uni[r], Bv);
    atomicAdd(&cnt[r], C);
  }
}

// ---------- kernel 4: per-sample dice reduction ----------
__global__ __launch_bounds__(256)
void dice_kernel(const float* __restrict__ inter, const float* __restrict__ uni,
                 const float* __restrict__ cnt, float* __restrict__ partial)
{
  __shared__ float s_d[256], s_n[256];
  int tid = threadIdx.x;
  int i = blockIdx.x * 256 + tid;
  float d = 0.0f, n = 0.0f;
  float c = cnt[i];
  if (c > 0.0f) {
    d = (2.0f * inter[i] + EPSF) / (uni[i] + EPSF);
    n = 1.0f;
  }
  s_d[tid] = d; s_n[tid] = n;
  __syncthreads();
  for (int s = 128; s > 0; s >>= 1) {
    if (tid < s) { s_d[tid] += s_d[tid + s]; s_n[tid] += s_n[tid + s]; }
    __syncthreads();
  }
  if (tid == 0 && s_n[0] > 0.0f) {
    atomicAdd(&partial[0], s_d[0]);
    atomicAdd(&partial[1], s_n[0]);
  }
}

// ---------- small helpers ----------
__global__ void zero_kernel(float* __restrict__ p, int n) {
  int i = blockIdx.x * blockDim.x + threadIdx.x;
  if (i < n) p[i] = 0.0f;
}

__global__ void final_kernel(const float* __restrict__ partial, float* __restrict__ out) {
  if (blockIdx.x == 0 && threadIdx.x == 0) {
    float acc = 0.0f;
    for (int b = 0; b < Bc; ++b) {
      float s = partial[2 * b], n = partial[2 * b + 1];
      acc += (n > 0.0f) ? (1.0f - s / n) : 1.0f;
    }
    out[0] = acc / (float)Bc;
  }
}

extern "C" void kernel_launch(void* const* d_in, const int* in_sizes, int n_in,
                              void* d_out, int out_size, void* d_ws, size_t ws_size,
                              hipStream_t stream)
{
  (void)in_sizes; (void)n_in; (void)out_size; (void)ws_size;
  const float* pred   = (const float*)d_in[0];
  const float* target = (const float*)d_in[1];
  float* out = (float*)d_out;

  // workspace layout (~17.1 MB, reused across samples via stream ordering)
  char* ws = (char*)d_ws;
  float* partial = (float*)ws; // 2 floats per sample
  size_t off = 256;
  int* L = (int*)(ws + off);                      off += (size_t)Nc * sizeof(int);
  unsigned char* Mk = (unsigned char*)(ws + off); off += (size_t)Nc;
  float* inter = (float*)(ws + off);              off += (size_t)Nc * sizeof(float);
  float* uni   = (float*)(ws + off);              off += (size_t)Nc * sizeof(float);
  float* cnt   = (float*)(ws + off);              off += (size_t)Nc * sizeof(float);

  zero_kernel<<<1, 32, 0, stream>>>(partial, 2 * Bc);

  const int P = (TILES_X - 1) * Hc + (TILES_Y - 1) * Wc; // boundary pairs
  for (int b = 0; b < Bc; ++b) {
    const float* pb = pred + (size_t)b * Nc;
    const float* tb = target + (size_t)b * Nc;
    tile_ccl_kernel<<<TILES_X * TILES_Y, TPB, 0, stream>>>(pb, tb, L, Mk, inter, uni, cnt);
    merge_kernel<<<(P + 255) / 256, 256, 0, stream>>>(L, Mk);
    accum_kernel<<<Nc / 256, 256, 0, stream>>>(pb, tb, L, Mk, inter, uni, cnt);
    dice_kernel<<<Nc / 256, 256, 0, stream>>>(inter, uni, cnt, partial + 2 * b);
  }
  final_kernel<<<1, 32, 0, stream>>>(partial, out);
}